// EdgeEncoder_25632364822642
// MI455X (gfx1250) — compile-verified
//
#include <hip/hip_runtime.h>

#define KNBRS 32
#define LDIM 10

typedef __attribute__((ext_vector_type(2))) float v2f;
typedef __attribute__((ext_vector_type(8))) float v8f;

// ---------------------------------------------------------------------------
// Kernel 0: pack (cx, cy, cx^2+cy^2, 1) per node for coalesced b128 loads.
// ---------------------------------------------------------------------------
__global__ void pack_pos_kernel(const float* __restrict__ loc,
                                float4* __restrict__ pk, int n) {
    int i = blockIdx.x * blockDim.x + threadIdx.x;
    if (i < n) {
        float cx = loc[i * LDIM + 6];
        float cy = loc[i * LDIM + 7];
        pk[i] = make_float4(cx, cy, cx * cx + cy * cy, 1.0f);
    }
}

// ---------------------------------------------------------------------------
// Kernel 1: KNN via V_WMMA_F32_16X16X4_F32.
//   One wave (32 threads) owns 16 target rows.  A-row m = (-2x_m, -2y_m,
//   |p_m|^2, 1); B-col n = (x_n, y_n, 1, |p_n|^2)  =>  D[m][n] = d^2(m,n)
//   exactly.  Per-row sorted top-32 lists kept in LDS; the rare insertion
//   path is ballot-gated and kept as a compact dynamic loop (no unroll) to
//   avoid exec-mask code bloat.
//
//   f32 16x4 A layout (ISA 7.12.2): lanes 0-15 hold K=0,1 of row M=lane;
//   lanes 16-31 hold K=2,3 of row M=lane-16.  B (4x16) mirrored.
//   C/D v8f: VGPR r = rows {r, r+8}, N = lane%16.
// ---------------------------------------------------------------------------
__global__ __launch_bounds__(32) void knn_wmma_kernel(
        const float4* __restrict__ pk, float* __restrict__ out_src, int n) {
    __shared__ float sd[16][KNBRS + 1];   // +1 pad: avoid row-aligned banks
    __shared__ int   si[16][KNBRS + 1];

    const int lane = threadIdx.x;         // 0..31 (wave32)
    const int m0   = blockIdx.x * 16;     // first target row of this tile
    const int hl   = lane & 15;           // lane-in-half

    // init top-k lists to +inf
    for (int m = 0; m < 16; ++m) {
        sd[m][lane] = __builtin_inff();
        si[m][lane] = 0;
    }
    __syncthreads();

    // A operand: fixed for the whole sweep.
    float4 pm = pk[m0 + hl];
    v2f a;
    if (lane < 16) { a.x = -2.0f * pm.x; a.y = -2.0f * pm.y; }  // K0,K1
    else           { a.x = pm.z;         a.y = 1.0f;         }  // K2,K3

    for (int base = 0; base < n; base += 16) {
        // B operand for this 16-source chunk.
        float4 pn = pk[base + hl];
        v2f b;
        if (lane < 16) { b.x = pn.x;  b.y = pn.y;  }  // K0,K1
        else           { b.x = 1.0f;  b.y = pn.z;  }  // K2,K3

        v8f c = {};
        v8f d = __builtin_amdgcn_wmma_f32_16x16x4_f32(
            /*neg_a=*/false, a, /*neg_b=*/false, b,
            /*c_mod=*/(short)0, c, /*reuse_a=*/false, /*reuse_b=*/false);

        const int ncol = base + hl;       // global source index this lane holds
#pragma unroll
        for (int r = 0; r < 8; ++r) {
            const int m  = (lane < 16) ? r : (r + 8);
            float dist   = d[r];
            if (m0 + m == ncol) dist = __builtin_inff();   // no self loop

            // cheap wave-wide early out: any lane beating its row's worst?
            unsigned long long any = __ballot(dist < sd[m][KNBRS - 1]);
            if (any) {
                // serialize within each half-wave; lane t (row r) and lane
                // 16+t (row r+8) insert into distinct LDS rows concurrently.
                // Kept as a dynamic loop: this path is rare, keep it small.
#pragma unroll 1
                for (int t = 0; t < 16; ++t) {
                    if (hl == t && dist < sd[m][KNBRS - 1]) {
                        int pos = KNBRS - 1;
#pragma unroll 1
                        while (pos > 0 && sd[m][pos - 1] > dist) {
                            sd[m][pos] = sd[m][pos - 1];
                            si[m][pos] = si[m][pos - 1];
                            --pos;
                        }
                        sd[m][pos] = dist;
                        si[m][pos] = ncol;
                    }
                }
            }
        }
    }
    __syncthreads();

    // emit src indices as float: out[(m0+m)*K + j] = nbr index j of row m0+m
    for (int m = 0; m < 16; ++m)
        out_src[(m0 + m) * KNBRS + lane] = (float)si[m][lane];
}

// ---------------------------------------------------------------------------
// Kernel 2: tgt row of edge_index + 4 edge features per edge.
//   Reads src ids back from out[0..E) (written by kernel 1, same stream).
// ---------------------------------------------------------------------------
__global__ void edge_attr_kernel(const float* __restrict__ loc,
                                 float* __restrict__ out, int n) {
    const int E = n * KNBRS;
    int e = blockIdx.x * blockDim.x + threadIdx.x;
    if (e >= E) return;

    const int tgt = e >> 5;               // e / KNBRS
    const int src = (int)out[e];

    const float* s = loc + src * LDIM;
    const float* t = loc + tgt * LDIM;

    float s4 = s[4], s5 = s[5], s6 = s[6], s7 = s[7];
    float t4 = t[4], t5 = t[5], t6 = t[6], t7 = t[7];

    float inv = 2.0f / (s5 + t5);
    float f1  = (s6 - t6) * inv;
    float f2  = (s7 - t7) * inv;
    float f3  = __logf(s4 / t4);
    float f4  = __logf(s5 / t5);

    out[E + e] = (float)tgt;              // edge_index row 1
    float* ea  = out + 2 * E + e * 4;     // edge_attr [E,4]
    ea[0] = f1; ea[1] = f2; ea[2] = f3; ea[3] = f4;
}

// ---------------------------------------------------------------------------
extern "C" void kernel_launch(void* const* d_in, const int* in_sizes, int n_in,
                              void* d_out, int out_size, void* d_ws, size_t ws_size,
                              hipStream_t stream) {
    (void)n_in; (void)out_size; (void)ws_size;
    const float* loc = (const float*)d_in[1];          // [N,10] float32
    const int n = in_sizes[1] / LDIM;                  // N nodes
    float4* pk = (float4*)d_ws;                        // packed (x,y,sq,1)
    float* out = (float*)d_out;

    pack_pos_kernel<<<(n + 255) / 256, 256, 0, stream>>>(loc, pk, n);
    knn_wmma_kernel<<<n / 16, 32, 0, stream>>>(pk, out, n);

    const int E = n * KNBRS;
    edge_attr_kernel<<<(E + 255) / 256, 256, 0, stream>>>(loc, out, n);
}